// StackAugmentedRNN_1133871366695
// MI455X (gfx1250) — compile-verified
//
#include <hip/hip_runtime.h>
#include <hip/hip_bf16.h>
#include <math.h>

#define V_  128
#define H_  1024
#define SW_ 1024
#define SD_ 8192

typedef __attribute__((ext_vector_type(2))) float v2f;
typedef __attribute__((ext_vector_type(4))) float v4f;
typedef __attribute__((ext_vector_type(8))) float v8f;

// workspace layout (float offsets)
#define WS_CTRL   0       // 3 floats: a_push, a_pop, a_noop
#define WS_SINP   16      // stack_input[1024]
#define WS_RNNIN  2048    // rnn_in[2048] = [ x | new_stack[0] ]
#define WS_GH     4096    // gh[3072]
#define WS_GI     8192    // gi[3072]
#define WS_HNEW   12288   // h_new[1024]

// ---------------------------------------------------------------------------
// GEMV via V_WMMA_F32_16X16X4_F32.
// out[row] = act( dot(W[row, 0:K], vec) + bias[row] ), rows = blockIdx.x*16..+16
// Block = 256 threads (8 waves). Wave w covers K-slice [w*K/8, (w+1)*K/8).
// A = vec broadcast over M; B = W^T tile; result y[n] = D[0,n] = c[0], lanes 0..15.
// Four independent accumulators break the WMMA RAW-on-C chain (4-deep ILP in
// the matrix pipe); partials are summed scalar-wise, matching the K-split
// re-association already used across waves.
// ---------------------------------------------------------------------------
template <int ACT> // 0 = identity, 1 = tanh
__global__ void gemv_wmma(const float* __restrict__ W, const float* __restrict__ vec,
                          const float* __restrict__ bias, float* __restrict__ out, int K) {
    const int lane = threadIdx.x & 31;
    const int wave = threadIdx.x >> 5;
    const int row_base = blockIdx.x * 16;
    const int kslice = K >> 3;              // per-wave K (128 or 256 here, /16 ok)
    const int kbeg = wave * kslice;

    const int r  = row_base + (lane & 15);  // weight row this lane streams
    const int co = (lane >> 4) << 1;        // 0 for lanes 0-15, 2 for lanes 16-31
    const float* wp = W + (size_t)r * K + (size_t)(kbeg + co);
    const float* vp = vec + kbeg + co;

    v8f c0 = {}, c1 = {}, c2 = {}, c3 = {};
    for (int k = 0; k < kslice; k += 16) {
        if ((k & 63) == 0) __builtin_prefetch(wp + k + 256, 0, 1); // global_prefetch_b8
        v2f a0 = *(const v2f*)(vp + k);
        v2f b0 = *(const v2f*)(wp + k);
        v2f a1 = *(const v2f*)(vp + k + 4);
        v2f b1 = *(const v2f*)(wp + k + 4);
        v2f a2 = *(const v2f*)(vp + k + 8);
        v2f b2 = *(const v2f*)(wp + k + 8);
        v2f a3 = *(const v2f*)(vp + k + 12);
        v2f b3 = *(const v2f*)(wp + k + 12);
        c0 = __builtin_amdgcn_wmma_f32_16x16x4_f32(false, a0, false, b0, (short)0, c0, false, false);
        c1 = __builtin_amdgcn_wmma_f32_16x16x4_f32(false, a1, false, b1, (short)0, c1, false, false);
        c2 = __builtin_amdgcn_wmma_f32_16x16x4_f32(false, a2, false, b2, (short)0, c2, false, false);
        c3 = __builtin_amdgcn_wmma_f32_16x16x4_f32(false, a3, false, b3, (short)0, c3, false, false);
    }

    __shared__ float red[8][16];
    if (lane < 16) red[wave][lane] = c0[0] + c1[0] + c2[0] + c3[0]; // D[0,lane] partials
    __syncthreads();
    if (threadIdx.x < 16) {
        float s = 0.f;
        #pragma unroll
        for (int w = 0; w < 8; ++w) s += red[w][threadIdx.x];
        s += bias[row_base + threadIdx.x];
        if (ACT == 1) s = tanhf(s);
        out[row_base + threadIdx.x] = s;
    }
}

// ---------------------------------------------------------------------------
// Controls softmax (3 logits) + gather x = emb[inp] into rnn_in[0:H]
// ---------------------------------------------------------------------------
__global__ void ctrl_and_x(const int* __restrict__ inp, const float* __restrict__ hidden,
                           const float* __restrict__ emb, const float* __restrict__ W_ctrl,
                           const float* __restrict__ b_ctrl, float* ws) {
    if (blockIdx.x == 0) {
        __shared__ float lg[3];
        const int wave = threadIdx.x >> 5, lane = threadIdx.x & 31;
        if (wave < 3) {
            float s = 0.f;
            for (int i = lane; i < H_; i += 32) s += W_ctrl[wave * H_ + i] * hidden[i];
            #pragma unroll
            for (int off = 16; off > 0; off >>= 1) s += __shfl_down(s, off, 32);
            if (lane == 0) lg[wave] = s + b_ctrl[wave];
        }
        __syncthreads();
        if (threadIdx.x == 0) {
            float m = fmaxf(lg[0], fmaxf(lg[1], lg[2]));
            float e0 = expf(lg[0] - m), e1 = expf(lg[1] - m), e2 = expf(lg[2] - m);
            float inv = 1.f / (e0 + e1 + e2);
            ws[WS_CTRL + 0] = e0 * inv;   // a_push
            ws[WS_CTRL + 1] = e1 * inv;   // a_pop
            ws[WS_CTRL + 2] = e2 * inv;   // a_noop
        }
    } else {
        int i = (blockIdx.x - 1) * blockDim.x + threadIdx.x;
        if (i < H_) ws[WS_RNNIN + i] = emb[(size_t)inp[0] * H_ + i];
    }
}

// ---------------------------------------------------------------------------
// new_stack = a_noop*stack + a_push*up + a_pop*down (v4f per thread).
// Reads stay RT (up to 3x reuse, 32 MB input is L2-resident in 192 MB);
// the 32 MB output is never re-read on-chip -> non-temporal stores.
// Also exports new_stack[0] into rnn_in[H:2H].
// ---------------------------------------------------------------------------
__global__ void stack_update(const float* __restrict__ stack, float* __restrict__ new_stack,
                             float* ws) {
    const float a_push = ws[WS_CTRL + 0];
    const float a_pop  = ws[WS_CTRL + 1];
    const float a_noop = ws[WS_CTRL + 2];

    const int idx = blockIdx.x * blockDim.x + threadIdx.x;   // v4f index
    const int row_f4 = SW_ / 4;
    const int j = idx / row_f4;
    const int c = (idx - j * row_f4) * 4;

    v4f cur = *(const v4f*)(stack + (size_t)j * SW_ + c);
    v4f up  = (j == 0) ? *(const v4f*)(ws + WS_SINP + c)
                       : *(const v4f*)(stack + (size_t)(j - 1) * SW_ + c);
    v4f dn = {};
    if (j < SD_ - 1) dn = *(const v4f*)(stack + (size_t)(j + 1) * SW_ + c);

    v4f o = a_noop * cur + a_push * up + a_pop * dn;

    __builtin_nontemporal_store(o, (v4f*)(new_stack + (size_t)j * SW_ + c));
    if (j == 0) *(v4f*)(ws + WS_RNNIN + H_ + c) = o;
}

// ---------------------------------------------------------------------------
// GRU gates: gi/gh already include their biases.
// ---------------------------------------------------------------------------
__global__ void gru_gates(const float* __restrict__ hidden, float* ws,
                          float* __restrict__ hid_out) {
    const int i = blockIdx.x * blockDim.x + threadIdx.x;
    if (i >= H_) return;
    const float* gi = ws + WS_GI;
    const float* gh = ws + WS_GH;
    float r  = 1.f / (1.f + expf(-(gi[i] + gh[i])));
    float z  = 1.f / (1.f + expf(-(gi[H_ + i] + gh[H_ + i])));
    float n  = tanhf(gi[2 * H_ + i] + r * gh[2 * H_ + i]);
    float hn = (1.f - z) * n + z * hidden[i];
    hid_out[i] = hn;
    ws[WS_HNEW + i] = hn;
}

// ---------------------------------------------------------------------------
extern "C" void kernel_launch(void* const* d_in, const int* in_sizes, int n_in,
                              void* d_out, int out_size, void* d_ws, size_t ws_size,
                              hipStream_t stream) {
    (void)in_sizes; (void)n_in; (void)out_size; (void)ws_size;
    const int*   inp    = (const int*)  d_in[0];
    const float* hidden = (const float*)d_in[1];
    const float* stack  = (const float*)d_in[2];
    const float* emb    = (const float*)d_in[3];
    const float* W_ctrl = (const float*)d_in[4];
    const float* b_ctrl = (const float*)d_in[5];
    const float* W_sin  = (const float*)d_in[6];
    const float* b_sin  = (const float*)d_in[7];
    const float* W_ih   = (const float*)d_in[8];
    const float* b_ih   = (const float*)d_in[9];
    const float* W_hh   = (const float*)d_in[10];
    const float* b_hh   = (const float*)d_in[11];
    const float* W_dec  = (const float*)d_in[12];
    const float* b_dec  = (const float*)d_in[13];

    float* out       = (float*)d_out;        // [0, 128)   logits
    float* hid_out   = out + V_;             // [128, 1152) next_hidden
    float* stack_out = out + V_ + H_;        // new_stack (8192*1024)
    float* ws        = (float*)d_ws;

    // 1) controls softmax + x = emb[inp]
    ctrl_and_x<<<dim3(1 + H_ / 128), dim3(128), 0, stream>>>(inp, hidden, emb, W_ctrl, b_ctrl, ws);
    // 2) stack_input = tanh(W_sin @ h + b_sin)
    gemv_wmma<1><<<dim3(SW_ / 16), dim3(256), 0, stream>>>(W_sin, hidden, b_sin, ws + WS_SINP, H_);
    // 3) gh = W_hh @ h + b_hh
    gemv_wmma<0><<<dim3(3 * H_ / 16), dim3(256), 0, stream>>>(W_hh, hidden, b_hh, ws + WS_GH, H_);
    // 4) stack blend -> new_stack, and rnn_in[H:2H] = new_stack[0]
    stack_update<<<dim3((SD_ * SW_ / 4) / 256), dim3(256), 0, stream>>>(stack, stack_out, ws);
    // 5) gi = W_ih @ rnn_in + b_ih   (K = 2048)
    gemv_wmma<0><<<dim3(3 * H_ / 16), dim3(256), 0, stream>>>(W_ih, ws + WS_RNNIN, b_ih, ws + WS_GI, H_ + SW_);
    // 6) GRU gates -> h_new
    gru_gates<<<dim3(H_ / 256), dim3(256), 0, stream>>>(hidden, ws, hid_out);
    // 7) out = W_dec @ h_new + b_dec
    gemv_wmma<0><<<dim3(V_ / 16), dim3(256), 0, stream>>>(W_dec, ws + WS_HNEW, b_dec, out, H_);
}